// GenODE_75170517615278
// MI455X (gfx1250) — compile-verified
//
#include <hip/hip_runtime.h>
#include <hip/hip_bf16.h>

// ---------------------------------------------------------------------------
// MI455X (gfx1250) implementation of the GenODE reference.
// fp32 end-to-end (rtol=1e-7 in the reference) using V_WMMA_F32_16X16X4_F32,
// with Tensor Data Mover (TDM) staging of GEMM tiles into LDS.
// ---------------------------------------------------------------------------

typedef __attribute__((ext_vector_type(2))) float v2f;
typedef __attribute__((ext_vector_type(8))) float v8f;
typedef __attribute__((ext_vector_type(4))) unsigned int u32x4;
typedef __attribute__((ext_vector_type(4))) int i32x4;
typedef __attribute__((ext_vector_type(8))) int i32x8;

__device__ __forceinline__ v8f wmma_f32(v2f a, v2f b, v8f c) {
  // (neg_a, A, neg_b, B, c_mod, C, reuse_a, reuse_b)
  return __builtin_amdgcn_wmma_f32_16x16x4_f32(false, a, false, b, (short)0, c,
                                               false, false);
}

// ===========================================================================
// TDM: DMA a 2D fp32 tile  global[valid_d1 x valid_d0, row stride `stride`]
// into LDS as a dense [tile_d1 x tile_d0] tile (OOB reads return zero).
// Issued by a single wave; caller does tdm_wait() + __syncthreads().
// Arity differs between toolchains (5-arg ROCm7.2 vs 6-arg therock): select
// via __has_include of the therock TDM header; fall back to a cooperative
// copy if the builtin does not exist at all.
// ===========================================================================
#if __has_include(<hip/amd_detail/amd_gfx1250_TDM.h>)
#define ATH_TDM_6ARG 1
#endif

#if defined(__has_builtin)
#if __has_builtin(__builtin_amdgcn_tensor_load_to_lds)
#define ATH_HAS_TDM 1
#endif
#endif

__device__ __forceinline__ void tdm_load_2d(const float* gsrc, float* ldst,
                                            unsigned valid_d0, unsigned valid_d1,
                                            unsigned tile_d0, unsigned tile_d1,
                                            unsigned stride) {
#if defined(ATH_HAS_TDM)
  const unsigned long long ga = (unsigned long long)(uintptr_t)gsrc;
  const unsigned lds_off = (unsigned)(uintptr_t)ldst;  // low 32 = LDS byte addr
  u32x4 g0;
  g0.x = 1u;                                       // count=1, no gather
  g0.y = lds_off;                                  // lds_addr
  g0.z = (unsigned)ga;                             // global_addr[31:0]
  g0.w = (unsigned)((ga >> 32) & 0x01FFFFFFu) | (2u << 30);  // addr[56:32],type=2
  i32x8 g1;
  g1[0] = (int)(2u << 16);                         // data_size = 4B
  g1[1] = (int)((valid_d0 & 0xFFFFu) << 16);       // tensor_dim0[15:0]
  g1[2] = (int)((valid_d0 >> 16) | ((valid_d1 & 0xFFFFu) << 16));
  g1[3] = (int)((valid_d1 >> 16) | (tile_d0 << 16));  // tensor_dim1 hi / tile_dim0
  g1[4] = (int)(tile_d1 & 0xFFFFu);                // tile_dim1 (tile_dim2=0)
  g1[5] = (int)stride;                             // tensor_dim0_stride[31:0]
  g1[6] = 0;                                       // stride hi / dim1_stride lo
  g1[7] = 0;
  const i32x4 z4 = {0, 0, 0, 0};
#if defined(ATH_TDM_6ARG)
  const i32x8 z8 = {0, 0, 0, 0, 0, 0, 0, 0};
  __builtin_amdgcn_tensor_load_to_lds(g0, g1, z4, z4, z8, 0);
#else
  __builtin_amdgcn_tensor_load_to_lds(g0, g1, z4, z4, 0);
#endif
#else
  // Fallback: one-wave cooperative copy with zero fill.
  const int lane = threadIdx.x & 31;
  for (unsigned i = lane; i < tile_d0 * tile_d1; i += 32) {
    unsigned r = i / tile_d0, c = i % tile_d0;
    ldst[r * tile_d0 + c] =
        (c < valid_d0 && r < valid_d1) ? gsrc[(size_t)r * stride + c] : 0.0f;
  }
#endif
}

__device__ __forceinline__ void tdm_wait() {
#if defined(ATH_HAS_TDM)
  __builtin_amdgcn_s_wait_tensorcnt(0);
#endif
}

// ===========================================================================
// Generic fused GEMM: C[M,N] = act(A[M,K] @ W[K,N] + bias[N]) (+ resid[M,N])
// Block = 128 threads = 4 waves; block tile 16(M)x64(N); each wave owns one
// 16x16 WMMA N-tile.  Both A and W tiles are TDM-staged into LDS per K-chunk
// (TDM zero-fills beyond tensor dims, so ragged K needs no special casing).
// ===========================================================================
#define KC 64

template <int ACT, int RES>
__global__ __launch_bounds__(128) void k_gemm(
    const float* __restrict__ A, const float* __restrict__ W,
    const float* __restrict__ bias, const float* __restrict__ resid,
    const float* __restrict__ alpha, float* __restrict__ C, int M, int N,
    int K) {
  __shared__ float sA[16 * KC];   // A chunk [16][KC]
  __shared__ float sW[KC * 64];   // W chunk [KC][64]
  const int tid = threadIdx.x;
  const int lane = tid & 31;
  const int wave = tid >> 5;
  const int m0 = blockIdx.y * 16;
  const int nb0 = blockIdx.x * 64;
  const int row = lane & 15;  // A row within tile / C column lane
  const int hil = lane >> 4;  // K-half select for A/B fragments

  v8f acc = {};
  const float a_val = (ACT == 1) ? alpha[0] : 0.0f;

  for (int kb = 0; kb < K; kb += KC) {
    __syncthreads();  // previous chunk fully consumed before DMA overwrite
    if (wave == 0) {
      const int krem = (K - kb < KC) ? (K - kb) : KC;
      tdm_load_2d(A + (size_t)m0 * K + kb, sA, (unsigned)krem, 16u, KC, 16u,
                  (unsigned)K);
      tdm_load_2d(W + (size_t)kb * N + nb0, sW, 64u, (unsigned)krem, 64u, KC,
                  (unsigned)N);
      tdm_wait();
    }
    __syncthreads();
#pragma unroll 4
    for (int k = 0; k < KC; k += 4) {
      // A fragment: lane holds A[row, kb+k+2*hil+{0,1}]  (ds_load_b64)
      v2f af = *(const v2f*)&sA[row * KC + k + 2 * hil];
      // B fragment: lane holds W[kb+k+2*hil+{0,1}, nb0+wave*16+row]
      const float* wr = &sW[(k + 2 * hil) * 64 + wave * 16 + row];
      v2f bf;
      bf.x = wr[0];
      bf.y = wr[64];
      acc = wmma_f32(af, bf, acc);
    }
  }

  // Epilogue: C layout -> lane row = v + 8*hil, col = nb0+wave*16+(lane&15)
  const int col = nb0 + wave * 16 + row;
  const float bv = bias[col];
#pragma unroll
  for (int v = 0; v < 8; ++v) {
    int r = m0 + v + 8 * hil;
    float x = acc[v] + bv;
    if (ACT == 1) x = (x >= 0.0f) ? x : a_val * x;
    if (RES) x += resid[(size_t)r * N + col];
    C[(size_t)r * N + col] = x;
  }
  (void)M;
}

// ===========================================================================
// ODE RHS MLP layer inside the integrator: out[:, :N] = in[:, :K] @ W + b
// in/out live in LDS ([16][264] buffers).  W streams straight from the 192MB
// L2 (the whole ODE weight set is ~1MB, resident after the first eval), which
// the scheduler overlaps with the WMMA chain.  K tail (K=129) via VALU.
// ===========================================================================
__device__ __forceinline__ void ode_layer(const float* __restrict__ W,
                                          const float* __restrict__ bias,
                                          int K, int N, const float (*in)[264],
                                          float (*out)[264], int tid) {
  const int lane = tid & 31;
  const int wave = tid >> 5;
  const int row = lane & 15;
  const int hil = lane >> 4;
  const int kmain = K & ~3;
  for (int n0 = wave * 16; n0 < N; n0 += 64) {
    v8f acc = {};
    for (int k = 0; k < kmain; k += 4) {
      v2f af = *(const v2f*)&in[row][k + 2 * hil];
      const float* wp = W + (size_t)(k + 2 * hil) * N + (n0 + row);
      if ((k & 31) == 0) __builtin_prefetch(wp + (size_t)32 * N, 0, 1);
      v2f bf;
      bf.x = wp[0];
      bf.y = wp[N];
      acc = wmma_f32(af, bf, acc);
    }
    const int col = n0 + row;
    for (int k = kmain; k < K; ++k) {  // K=129 tail (time column)
      float wv = W[(size_t)k * N + col];
#pragma unroll
      for (int v = 0; v < 8; ++v) acc[v] += in[v + 8 * hil][k] * wv;
    }
    const float bv = bias[col];
#pragma unroll
    for (int v = 0; v < 8; ++v) out[v + 8 * hil][col] = acc[v] + bv;
  }
}

// LayerNorm (population var, eps=1e-5) + PReLU:  bufB -> bufA   (N fixed 256)
__device__ __forceinline__ void ln_prelu(const float* __restrict__ g,
                                         const float* __restrict__ bt, float al,
                                         const float (*in)[264],
                                         float (*out)[264], float* smu,
                                         float* srs, int tid) {
  if (tid < 16) {
    float s = 0.0f, s2 = 0.0f;
    for (int c = 0; c < 256; ++c) {
      float x = in[tid][c];
      s += x;
      s2 += x * x;
    }
    float mu = s * (1.0f / 256.0f);
    float var = s2 * (1.0f / 256.0f) - mu * mu;
    smu[tid] = mu;
    srs[tid] = rsqrtf(var + 1e-5f);
  }
  __syncthreads();
  for (int i = tid; i < 16 * 256; i += 128) {
    int r = i >> 8, c = i & 255;
    float x = (in[r][c] - smu[r]) * srs[r] * g[c] + bt[c];
    out[r][c] = (x >= 0.0f) ? x : al * x;
  }
  __syncthreads();
}

struct OdeParams {
  const float* w[5];
  const float* b[5];
  const float* g[4];
  const float* bt[4];
  const float* al[4];
};

#define ODE_SUB 8  // fixed RK4 substeps per output interval

// ===========================================================================
// Persistent integrator: each block owns 16 batch rows (independent ODEs),
// keeps state/accumulators/activations in LDS (~58 KB of the 320 KB WGP pool)
// and integrates all T output points without touching HBM between steps.
// ===========================================================================
__global__ __launch_bounds__(128) void k_ode(OdeParams P,
                                             const float* __restrict__ tarr,
                                             const float* __restrict__ h0,
                                             float* __restrict__ sol, int T) {
  __shared__ float bufA[16][264];  // activated layer input (incl. time col 128)
  __shared__ float bufB[16][264];  // raw layer output
  __shared__ float sy[16][128];    // state y
  __shared__ float skv[16][128];   // current RK stage k
  __shared__ float sacc[16][128];  // RK accumulator
  __shared__ float smu[16], srs[16];

  const int tid = threadIdx.x;
  const int wave = tid >> 5;
  const int m0 = blockIdx.x * 16;

  // TDM-stage the initial state tile, then publish sol[:, 0, :].
  if (wave == 0) {
    tdm_load_2d(h0 + (size_t)m0 * 128, &sy[0][0], 128u, 16u, 128u, 16u, 128u);
    tdm_wait();
  }
  __syncthreads();
  for (int i = tid; i < 16 * 128; i += 128) {
    int r = i >> 7, c = i & 127;
    sol[((size_t)(m0 + r) * T) * 128 + c] = sy[r][c];
    skv[r][c] = 0.0f;
  }
  __syncthreads();

  for (int j = 0; j < T - 1; ++j) {
    const float t0 = tarr[j];
    const float hstep = (tarr[j + 1] - t0) / (float)ODE_SUB;
    for (int s = 0; s < ODE_SUB; ++s) {
      const float tcur = t0 + hstep * (float)s;
      for (int i = tid; i < 16 * 128; i += 128) {
        int r = i >> 7, c = i & 127;
        sacc[r][c] = 0.0f;
        skv[r][c] = 0.0f;
      }
      __syncthreads();
      for (int e = 0; e < 4; ++e) {
        const float ce = (e == 0) ? 0.0f : ((e == 3) ? 1.0f : 0.5f);
        const float wt = (e == 1 || e == 2) ? 2.0f : 1.0f;
        const float cc = ce * hstep;
        const float te = tcur + ce * hstep;
        // build RHS input  [y + c*h*k_prev , t]
        for (int i = tid; i < 16 * 128; i += 128) {
          int r = i >> 7, c = i & 127;
          bufA[r][c] = sy[r][c] + cc * skv[r][c];
        }
        if (tid < 16) bufA[tid][128] = te;
        __syncthreads();

        ode_layer(P.w[0], P.b[0], 129, 256, bufA, bufB, tid);
        __syncthreads();
        ln_prelu(P.g[0], P.bt[0], P.al[0][0], bufB, bufA, smu, srs, tid);
        for (int li = 1; li <= 3; ++li) {
          ode_layer(P.w[li], P.b[li], 256, 256, bufA, bufB, tid);
          __syncthreads();
          ln_prelu(P.g[li], P.bt[li], P.al[li][0], bufB, bufA, smu, srs, tid);
        }
        ode_layer(P.w[4], P.b[4], 256, 128, bufA, bufB, tid);
        __syncthreads();

        for (int i = tid; i < 16 * 128; i += 128) {
          int r = i >> 7, c = i & 127;
          float kv = bufB[r][c];
          skv[r][c] = kv;
          sacc[r][c] += wt * kv;
        }
        __syncthreads();
      }
      const float sc = hstep * (1.0f / 6.0f);
      for (int i = tid; i < 16 * 128; i += 128) {
        int r = i >> 7, c = i & 127;
        sy[r][c] += sc * sacc[r][c];
      }
      __syncthreads();
    }
    for (int i = tid; i < 16 * 128; i += 128) {
      int r = i >> 7, c = i & 127;
      sol[((size_t)(m0 + r) * T + (j + 1)) * 128 + c] = sy[r][c];
    }
  }
}

// ===========================================================================
// Fused decoder head: out[m] = prelu(Y[m,:]@W1 + b1) @ w2 + b2
// 16 rows/block; Y tile and per-pass W1 panel (128x64) are TDM-staged into
// LDS; the 1024-wide reduction goes through LDS float atomics (ds_add_f32).
// Fusing avoids materializing the 537 MB hidden tensor in HBM.
// ===========================================================================
__global__ __launch_bounds__(128) void k_dec_head(
    const float* __restrict__ Y, const float* __restrict__ W1,
    const float* __restrict__ b1, const float* __restrict__ alpha,
    const float* __restrict__ w2, const float* __restrict__ b2,
    float* __restrict__ out, int M) {
  __shared__ float sY[16 * 128];
  __shared__ float sW[128 * 64];  // W1 panel: 128 K-rows x 64 hidden cols
  __shared__ float sAcc[16];
  const int tid = threadIdx.x;
  const int lane = tid & 31;
  const int wave = tid >> 5;
  const int m0 = blockIdx.x * 16;
  const int row = lane & 15;
  const int hil = lane >> 4;

  if (wave == 0) {
    tdm_load_2d(Y + (size_t)m0 * 128, sY, 128u, 16u, 128u, 16u, 128u);
    tdm_wait();
  }
  if (tid < 16) sAcc[tid] = 0.0f;
  __syncthreads();

  const float av = alpha[0];
  for (int pass = 0; pass < 16; ++pass) {  // 16 * 64 = 1024 hidden cols
    __syncthreads();  // previous panel fully consumed
    if (wave == 0) {
      tdm_load_2d(W1 + pass * 64, sW, 64u, 128u, 64u, 128u, 1024u);
      tdm_wait();
    }
    __syncthreads();
    v8f acc = {};
#pragma unroll 4
    for (int k = 0; k < 128; k += 4) {
      v2f af = *(const v2f*)&sY[row * 128 + k + 2 * hil];
      const float* wr = &sW[(k + 2 * hil) * 64 + wave * 16 + row];
      v2f bf;
      bf.x = wr[0];
      bf.y = wr[64];
      acc = wmma_f32(af, bf, acc);
    }
    const int col = pass * 64 + wave * 16 + row;
    const float bb = b1[col];
    const float wv = w2[col];
#pragma unroll
    for (int v = 0; v < 8; ++v) {
      float h = acc[v] + bb;
      h = (h >= 0.0f) ? h : av * h;
      atomicAdd(&sAcc[v + 8 * hil], h * wv);  // ds_add_f32
    }
  }
  __syncthreads();
  if (tid < 16 && (m0 + tid) < M) out[m0 + tid] = sAcc[tid] + b2[0];
}

// ===========================================================================
// Host: d_in flattening order (setup_inputs dict order):
//  0 x[1024*8]  1 t[128]
//  aug: 2 l1.w  3 l1.b  4 a  5 l2.w  6 l2.b ; res[i] at 7+5i: w,b,a,w2,b2
//  ode: lins[i] at 22+2i: w,b ; ln[i] at 32+2i: g,b ; a[i] at 40+i
//  dec: res[i] at 44+5i: w,b,a,w2,b2 ; 59 l1.w 60 l1.b 61 a 62 l2.w 63 l2.b
// ===========================================================================
extern "C" void kernel_launch(void* const* d_in, const int* in_sizes, int n_in,
                              void* d_out, int out_size, void* d_ws,
                              size_t ws_size, hipStream_t stream) {
  (void)n_in;
  (void)out_size;
  (void)ws_size;
  const float* x = (const float*)d_in[0];
  const float* tt = (const float*)d_in[1];
  const int B = in_sizes[0] / 8;    // 1024
  const int T = in_sizes[1];        // 128
  const size_t BT = (size_t)B * T;  // 131072

  float* ws = (float*)d_ws;
  float* tmp = ws;                     // BT*128 floats (also aug hidden)
  float* h0 = tmp + BT * 128;          // B*128
  float* solA = h0 + (size_t)B * 128;  // BT*128
  float* solB = solA + BT * 128;       // BT*128

  const dim3 blk(128);

  // ---- aug head: prelu(x@W1+b1) @ W2 + b2 ----
  k_gemm<1, 0><<<dim3(1024 / 64, B / 16), blk, 0, stream>>>(
      x, (const float*)d_in[2], (const float*)d_in[3], nullptr,
      (const float*)d_in[4], tmp, B, 1024, 8);
  k_gemm<0, 0><<<dim3(128 / 64, B / 16), blk, 0, stream>>>(
      tmp, (const float*)d_in[5], (const float*)d_in[6], nullptr, nullptr, h0,
      B, 128, 1024);

  // ---- aug resblocks (in-place on h0) ----
  for (int i = 0; i < 3; ++i) {
    const int base = 7 + 5 * i;
    k_gemm<1, 0><<<dim3(2, B / 16), blk, 0, stream>>>(
        h0, (const float*)d_in[base], (const float*)d_in[base + 1], nullptr,
        (const float*)d_in[base + 2], tmp, B, 128, 128);
    k_gemm<0, 1><<<dim3(2, B / 16), blk, 0, stream>>>(
        tmp, (const float*)d_in[base + 3], (const float*)d_in[base + 4], h0,
        nullptr, h0, B, 128, 128);
  }

  // ---- ODE integration (persistent per-row-tile RK4) ----
  OdeParams P;
  for (int i = 0; i < 5; ++i) {
    P.w[i] = (const float*)d_in[22 + 2 * i];
    P.b[i] = (const float*)d_in[23 + 2 * i];
  }
  for (int i = 0; i < 4; ++i) {
    P.g[i] = (const float*)d_in[32 + 2 * i];
    P.bt[i] = (const float*)d_in[33 + 2 * i];
    P.al[i] = (const float*)d_in[40 + i];
  }
  k_ode<<<B / 16, blk, 0, stream>>>(P, tt, h0, solA, T);

  // ---- decoder resblocks on [B*T, 128] ----
  float* rin = solA;
  float* rout = solB;
  for (int i = 0; i < 3; ++i) {
    const int base = 44 + 5 * i;
    k_gemm<1, 0><<<dim3(2, (unsigned)(BT / 16)), blk, 0, stream>>>(
        rin, (const float*)d_in[base], (const float*)d_in[base + 1], nullptr,
        (const float*)d_in[base + 2], tmp, (int)BT, 128, 128);
    k_gemm<0, 1><<<dim3(2, (unsigned)(BT / 16)), blk, 0, stream>>>(
        tmp, (const float*)d_in[base + 3], (const float*)d_in[base + 4], rin,
        nullptr, rout, (int)BT, 128, 128);
    float* sw = rin;
    rin = rout;
    rout = sw;
  }

  // ---- fused decoder head ----
  k_dec_head<<<(unsigned)(BT / 16), blk, 0, stream>>>(
      rin, (const float*)d_in[59], (const float*)d_in[60],
      (const float*)d_in[61], (const float*)d_in[62], (const float*)d_in[63],
      (float*)d_out, (int)BT);
}